// SkilLoss_48473000903016
// MI455X (gfx1250) — compile-verified
//
#include <hip/hip_runtime.h>
#include <math.h>

// ---------------------------------------------------------------------------
// SKIL loss pipeline for MI455X (gfx1250, wave32).
// Stencil stages: LDS-tiled VALU min/max.  Gaussian blur: V_WMMA_F32_16X16X4_F32
// (fp32 matrix pipe, reference precision), conv-as-banded-GEMM.
// Interior tiles take a wave-uniform (scalar-branch) fast path with pure
// immediate-offset loads; boundary tiles use branchless clamp + cndmask.
// EXEC stays full everywhere a WMMA executes.
// ---------------------------------------------------------------------------

typedef __attribute__((ext_vector_type(2))) float v2f;
typedef __attribute__((ext_vector_type(8))) float v8f;

#define IMG_W 1024
#define IMG_H 1024
#define NBATCH 8
#define KPAD 68          // 51-tap window padded to 68 so K-loop is 17 x (K=4)
#define EPSF 1e-6f

// Branchless guarded load: clamp index into range, load unconditionally,
// select 0 if the unclamped index was out of bounds (v_cndmask, no branch).
__device__ __forceinline__ float ldg_row(const float* __restrict__ img,
                                         int gy, int col) {
  int gyc = gy < 0 ? 0 : (gy > IMG_H - 1 ? IMG_H - 1 : gy);
  float v = img[(size_t)gyc * IMG_W + col];
  return (gy == gyc) ? v : 0.0f;
}
__device__ __forceinline__ float ldg_col(const float* __restrict__ img,
                                         int row, int gx) {
  int gxc = gx < 0 ? 0 : (gx > IMG_W - 1 ? IMG_W - 1 : gx);
  float v = img[(size_t)row * IMG_W + gxc];
  return (gx == gxc) ? v : 0.0f;
}

// ---------------- setup: gaussian band table (zero-padded) + accumulators ---
// gtab[j] = gauss(j-32-25) for (j-32) in [0,50], else 0.  Band offset d=k-m
// (range [-15,67]) is looked up as gtab[d+32] with no branches.
__global__ void setup_kernel(float* __restrict__ gtab, float* __restrict__ acc) {
  int j = threadIdx.x;                 // 128 threads
  float v = 0.0f;
  int d = j - 32;
  if (d >= 0 && d <= 50) {
    float t = (float)(d - 25);
    v = expf(-(t * t) / (2.0f * 10.0f * 10.0f));
  }
  gtab[j] = v;
  if (j < 24) acc[j] = 0.0f;           // [0..7]=gp  [8..15]=gg  [16..23]=pp
}

// ---------------- softmax foreground channel -------------------------------
__global__ __launch_bounds__(256) void softmax_fg_kernel(
    const float* __restrict__ pred, float* __restrict__ fg) {
  size_t i = (size_t)blockIdx.x * blockDim.x + threadIdx.x;   // 8M threads
  const size_t M = (size_t)IMG_W * IMG_H;
  size_t b = i / M, pix = i - b * M;
  float p0 = pred[b * 2 * M + pix];
  float p1 = pred[b * 2 * M + M + pix];
  fg[i] = 1.0f / (1.0f + expf(p0 - p1));   // softmax over 2 channels
}

// ---------------- fused skeleton step --------------------------------------
// One pass of:  e = erode(x);  d = dilate(e);  delta = relu(x - d);
//               skel = first ? delta : skel + relu(delta - skel*delta);
//               xout = e;
// erode OOB neighbors -> +inf (ignored by min), dilate OOB e-centers -> -inf.
__global__ __launch_bounds__(256) void skel_step_kernel(
    const float* __restrict__ xin, float* __restrict__ skel,
    float* __restrict__ xout, int first) {
  __shared__ float sx[12][36];   // x tile + halo 2
  __shared__ float se[12][36];   // erode tile + halo 1
  const int W = IMG_W, H = IMG_H;
  int b  = blockIdx.z;
  int x0 = blockIdx.x * 32;
  int y0 = blockIdx.y * 8;
  const float* img = xin + (size_t)b * W * H;
  int tid = threadIdx.y * 32 + threadIdx.x;

  for (int i = tid; i < 36 * 12; i += 256) {
    int r = i / 36, c = i % 36;
    int gy = y0 + r - 2, gx = x0 + c - 2;
    bool inb = (gy >= 0 && gy < H && gx >= 0 && gx < W);
    int gyc = gy < 0 ? 0 : (gy > H - 1 ? H - 1 : gy);
    int gxc = gx < 0 ? 0 : (gx > W - 1 ? W - 1 : gx);
    float v = img[(size_t)gyc * W + gxc];
    sx[r][c] = inb ? v : __builtin_inff();            // +inf: ignored by min
  }
  __syncthreads();

  for (int i = tid; i < 34 * 10; i += 256) {
    int r = i / 34 + 1, c = i % 34 + 1;
    int gy = y0 + r - 2, gx = x0 + c - 2;
    bool inb = (gy >= 0 && gy < H && gx >= 0 && gx < W);
    float e = sx[r][c];
    e = fminf(e, sx[r][c - 1]);     e = fminf(e, sx[r][c + 1]);
    e = fminf(e, sx[r - 1][c - 1]); e = fminf(e, sx[r - 1][c]); e = fminf(e, sx[r - 1][c + 1]);
    e = fminf(e, sx[r + 1][c - 1]); e = fminf(e, sx[r + 1][c]); e = fminf(e, sx[r + 1][c + 1]);
    se[r][c] = inb ? e : -__builtin_inff();           // OOB centers ignored by max
  }
  __syncthreads();

  int r = threadIdx.y + 2, c = threadIdx.x + 2;
  int gy = y0 + threadIdx.y, gx = x0 + threadIdx.x;
  float d = se[r][c];
  d = fmaxf(d, se[r][c - 1]);     d = fmaxf(d, se[r][c + 1]);
  d = fmaxf(d, se[r - 1][c - 1]); d = fmaxf(d, se[r - 1][c]); d = fmaxf(d, se[r - 1][c + 1]);
  d = fmaxf(d, se[r + 1][c - 1]); d = fmaxf(d, se[r + 1][c]); d = fmaxf(d, se[r + 1][c + 1]);

  size_t idx = (size_t)b * W * H + (size_t)gy * W + gx;
  float delta = fmaxf(sx[r][c] - d, 0.0f);
  if (first) skel[idx] = delta;
  else {
    float s = skel[idx];
    skel[idx] = s + fmaxf(delta - s * delta, 0.0f);
  }
  xout[idx] = se[r][c];
}

// ---------------- WMMA vertical blur:  O = Gband(16xK) x ImgTile(Kx16) -----
// A (const band):   A[m][k] = gauss[k-m],  a[v] = gtab[kb+2*half+v - m + 32]
// B (image tile):   b[v]    = img[y0-25 + kb+2*half+v][x0+n] (0 outside)
__global__ __launch_bounds__(256) void blur_v_kernel(
    const float* __restrict__ src, float* __restrict__ dst,
    const float* __restrict__ gtab) {
  const int W = IMG_W, H = IMG_H;
  int lane = threadIdx.x & 31, wave = threadIdx.x >> 5;
  int n = lane & 15, hp = lane >> 4;
  int b  = blockIdx.z;
  int x0 = (blockIdx.x * 8 + wave) * 16;
  int y0 = blockIdx.y * 16;                      // pure blockIdx.y -> SGPR
  const float* img = src + (size_t)b * W * H;
  v8f c = {0.f, 0.f, 0.f, 0.f, 0.f, 0.f, 0.f, 0.f};
  const float* gt = gtab + 32 + 2 * hp - n;      // per-lane gauss base

  if (y0 >= 25 && y0 - 25 + KPAD <= IMG_H) {     // scalar branch: interior tile
    const float* p = img + (size_t)(y0 - 25 + 2 * hp) * W + x0 + n;
#pragma unroll
    for (int kb = 0; kb < KPAD; kb += 4) {
      v2f a, bm;
      a.x  = gt[kb + 0];
      a.y  = gt[kb + 1];
      bm.x = p[(size_t)kb * W];                  // immediate offset kb*4096
      bm.y = p[(size_t)(kb + 1) * W];
      c = __builtin_amdgcn_wmma_f32_16x16x4_f32(false, a, false, bm,
                                                (short)0, c, false, false);
    }
  } else {                                       // boundary tile: clamp+cndmask
#pragma unroll
    for (int kb = 0; kb < KPAD; kb += 4) {
      int k0 = kb + 2 * hp;
      v2f a, bm;
      a.x  = gt[kb + 0];
      a.y  = gt[kb + 1];
      int gy0 = y0 - 25 + k0;
      bm.x = ldg_row(img, gy0, x0 + n);
      bm.y = ldg_row(img, gy0 + 1, x0 + n);
      c = __builtin_amdgcn_wmma_f32_16x16x4_f32(false, a, false, bm,
                                                (short)0, c, false, false);
    }
  }
  float* out = dst + (size_t)b * W * H;
#pragma unroll
  for (int v = 0; v < 8; ++v) {
    int row = y0 + v + 8 * hp;
    out[(size_t)row * W + x0 + n] = c[v];
  }
}

// ---------------- WMMA horizontal blur + clip (+ dice partials) ------------
// A (image tile): a[v] = img[y0+m][x0-25 + kb+2*half+v] (0 outside)
// B (const band): b[v] = gtab[kb+2*half+v - n + 32]
// MODE=0: store gt_border, accumulate sum(gt^2).
// MODE=1: read gt_border, accumulate sum(gt*pr), sum(pr^2).
template <int MODE>
__global__ __launch_bounds__(256) void blur_h_kernel(
    const float* __restrict__ src, float* __restrict__ gtb,
    const float* __restrict__ gtab, float* __restrict__ acc) {
  const int W = IMG_W, H = IMG_H;
  int lane = threadIdx.x & 31;
  // wave id is wave-uniform; force into SGPR so the interior test is scalar
  int wave = __builtin_amdgcn_readfirstlane((int)(threadIdx.x >> 5));
  int n = lane & 15, hp = lane >> 4;
  int b  = blockIdx.z;
  int x0 = (blockIdx.x * 8 + wave) * 16;         // scalar
  int y0 = blockIdx.y * 16;
  const float* img = src + (size_t)b * W * H;
  v8f c = {0.f, 0.f, 0.f, 0.f, 0.f, 0.f, 0.f, 0.f};
  const float* gt = gtab + 32 + 2 * hp - n;      // per-lane gauss base

  if (x0 >= 25 && x0 - 25 + KPAD <= IMG_W) {     // scalar branch: interior tile
    const float* p = img + (size_t)(y0 + n) * W + x0 - 25 + 2 * hp;
#pragma unroll
    for (int kb = 0; kb < KPAD; kb += 4) {
      v2f a, bm;
      a.x  = p[kb];                              // immediate offset kb*4
      a.y  = p[kb + 1];
      bm.x = gt[kb + 0];
      bm.y = gt[kb + 1];
      c = __builtin_amdgcn_wmma_f32_16x16x4_f32(false, a, false, bm,
                                                (short)0, c, false, false);
    }
  } else {                                       // boundary tile: clamp+cndmask
#pragma unroll
    for (int kb = 0; kb < KPAD; kb += 4) {
      int k0 = kb + 2 * hp;
      int gx0 = x0 - 25 + k0;
      v2f a, bm;
      a.x  = ldg_col(img, y0 + n, gx0);
      a.y  = ldg_col(img, y0 + n, gx0 + 1);
      bm.x = gt[kb + 0];
      bm.y = gt[kb + 1];
      c = __builtin_amdgcn_wmma_f32_16x16x4_f32(false, a, false, bm,
                                                (short)0, c, false, false);
    }
  }

  float s0 = 0.0f, s1 = 0.0f;
  float* gimg = gtb + (size_t)b * W * H;
#pragma unroll
  for (int v = 0; v < 8; ++v) {
    int row = y0 + v + 8 * hp;
    size_t idx = (size_t)row * W + x0 + n;
    float val = fminf(fmaxf(c[v], 0.0f), 1.0f);   // clip(y, 0, 1)
    if (MODE == 0) {
      gimg[idx] = val;
      s0 += val * val;                            // gg
    } else {
      float g = gimg[idx];
      s0 += g * val;                              // gp
      s1 += val * val;                            // pp
    }
  }
  for (int off = 16; off > 0; off >>= 1) {        // wave32 reduce
    s0 += __shfl_xor(s0, off, 32);
    if (MODE == 1) s1 += __shfl_xor(s1, off, 32);
  }
  if (lane == 0) {
    if (MODE == 0) atomicAdd(&acc[8 + b], s0);
    else { atomicAdd(&acc[0 + b], s0); atomicAdd(&acc[16 + b], s1); }
  }
}

// ---------------- finalize: mean over batch of (1 - dice) ------------------
__global__ void finalize_kernel(const float* __restrict__ acc,
                                float* __restrict__ out) {
  if (threadIdx.x == 0 && blockIdx.x == 0) {
    float loss = 0.0f;
    for (int b = 0; b < NBATCH; ++b) {
      float num = 2.0f * acc[b] + EPSF;
      float den = acc[8 + b] + acc[16 + b] + EPSF;
      loss += 1.0f - num / den;
    }
    out[0] = loss * (1.0f / (float)NBATCH);      // LOSS_WEIGHT = 1
  }
}

// ---------------------------------------------------------------------------
extern "C" void kernel_launch(void* const* d_in, const int* in_sizes, int n_in,
                              void* d_out, int out_size, void* d_ws, size_t ws_size,
                              hipStream_t stream) {
  (void)in_sizes; (void)n_in; (void)out_size; (void)ws_size;
  const float* pred   = (const float*)d_in[0];   // (8,2,1024,1024) f32
  const float* target = (const float*)d_in[1];   // (8,1024,1024)  f32
  float* out = (float*)d_out;

  float* gtab = (float*)d_ws;                    // 128 floats
  float* acc  = gtab + 128;                      // 24 floats
  float* base = gtab + 256;                      // big buffers
  const size_t IMG = (size_t)NBATCH * IMG_W * IMG_H;   // 8M floats = 32 MB
  float* bufA = base;
  float* bufB = base + IMG;
  float* skel = base + 2 * IMG;
  float* gtb  = base + 3 * IMG;                  // gt_border

  setup_kernel<<<1, 128, 0, stream>>>(gtab, acc);

  dim3 sblk(32, 8, 1), sgrd(IMG_W / 32, IMG_H / 8, NBATCH);
  dim3 bblk(256, 1, 1), bgrd(IMG_W / 128, IMG_H / 16, NBATCH);

  // ---- target field: skeletonize (11 fused passes), blur, store gt border
  {
    const float* xin = target; float* xout = bufA;
    for (int k = 0; k <= 10; ++k) {
      skel_step_kernel<<<sgrd, sblk, 0, stream>>>(xin, skel, xout, k == 0 ? 1 : 0);
      if (k == 0) { xin = bufA; xout = bufB; }
      else        { float* t = (float*)xin; xin = xout; xout = t; }
    }
  }
  blur_v_kernel<<<bgrd, bblk, 0, stream>>>(skel, bufB, gtab);
  blur_h_kernel<0><<<bgrd, bblk, 0, stream>>>(bufB, gtb, gtab, acc);

  // ---- prediction field: softmax fg, skeletonize, blur, fused dice sums
  softmax_fg_kernel<<<(unsigned)(IMG / 256), 256, 0, stream>>>(pred, bufA);
  {
    const float* xin = bufA; float* xout = bufB;
    for (int k = 0; k <= 10; ++k) {
      skel_step_kernel<<<sgrd, sblk, 0, stream>>>(xin, skel, xout, k == 0 ? 1 : 0);
      float* t = (float*)xin; xin = xout; xout = t;
    }
  }
  blur_v_kernel<<<bgrd, bblk, 0, stream>>>(skel, bufA, gtab);
  blur_h_kernel<1><<<bgrd, bblk, 0, stream>>>(bufA, gtb, gtab, acc);

  finalize_kernel<<<1, 32, 0, stream>>>(acc, out);
}